// HBS_40346922779263
// MI455X (gfx1250) — compile-verified
//
#include <hip/hip_runtime.h>

#define N_CELLS 50000
#define N_EDGES 1600000
#define D_IN 256
#define D_OUT 128
#define NEG_SLOPE 0.2f

#define KCHUNK 64                                  // K-rows of W per LDS chunk
#define CHUNK_BYTES (KCHUNK * D_OUT * 4)           // 32 KB
#define NCHUNK (D_IN / KCHUNK)                     // 4

typedef __attribute__((ext_vector_type(2))) float v2f;
typedef __attribute__((ext_vector_type(8))) float v8f;

// ---------------------------------------------------------------------------
// gfx1250 async-to-LDS helpers (ASYNCcnt-tracked DMA, bypasses VGPRs)
// ---------------------------------------------------------------------------
__device__ __forceinline__ void async_stage_chunk(const float* __restrict__ gsrc,
                                                  unsigned lds_byte_off) {
    // Copy CHUNK_BYTES (32 KB): 256 threads x 8 iters x 16 B, per-lane DMA.
    const unsigned t = threadIdx.x * 16u;
#pragma unroll
    for (int it = 0; it < CHUNK_BYTES / (256 * 16); ++it) {   // 8
        const unsigned off = t + it * 4096u;
        asm volatile("global_load_async_to_lds_b128 %0, %1, off"
                     :: "v"(lds_byte_off + off),
                        "v"((const char*)gsrc + off)
                     : "memory");
    }
}

__device__ __forceinline__ void wait_async0() {
#if __has_builtin(__builtin_amdgcn_s_wait_asynccnt)
    __builtin_amdgcn_s_wait_asynccnt(0);
#else
    asm volatile("s_wait_asynccnt 0x0" ::: "memory");
#endif
}

// ---------------------------------------------------------------------------
// Phase 0: zero a float buffer (d_out and row_sum must be zeroed each launch)
// ---------------------------------------------------------------------------
__global__ void hbs_zero(float* __restrict__ p, int n) {
    int i = blockIdx.x * blockDim.x + threadIdx.x;
    if (i < n) p[i] = 0.0f;
}

// ---------------------------------------------------------------------------
// Phase 1: m = x @ W via f32 WMMA, W staged through LDS with double-buffered
// async-to-LDS DMA. One wave per 16x16 output tile; 8 waves cover D_OUT.
// grid.x = N_CELLS/16 = 3125 (exact, EXEC all-1s for WMMA).
// ---------------------------------------------------------------------------
__global__ __launch_bounds__(256) void hbs_gemm(const float* __restrict__ x,
                                                const float* __restrict__ W,
                                                float* __restrict__ m) {
    extern __shared__ float lw[];        // 2 x (KCHUNK x D_OUT) floats = 64 KB
    const int lane = threadIdx.x & 31;
    const int wave = threadIdx.x >> 5;   // 0..7 -> N tile
    const int row0 = blockIdx.x * 16;
    const int col0 = wave * 16;
    const int l15  = lane & 15;
    const int hi   = lane >> 4;          // lane half selects K pair

    // Prologue: stage chunk 0
    async_stage_chunk(W, 0u);
    wait_async0();
    __syncthreads();

    v8f c = {};
    const float* xrow = x + (row0 + l15) * D_IN;   // A: row M = l15

    for (int kc = 0; kc < NCHUNK; ++kc) {
        if (kc + 1 < NCHUNK)
            async_stage_chunk(W + (kc + 1) * KCHUNK * D_OUT,
                              (unsigned)(((kc + 1) & 1) * CHUNK_BYTES));

        // B operand from LDS: col N = l15, K split across lane halves.
        const float* lb = lw + (kc & 1) * (KCHUNK * D_OUT) + col0 + l15;
        const float* xk = xrow + kc * KCHUNK;
#pragma unroll
        for (int kk = 0; kk < KCHUNK; kk += 4) {
            const int ka = kk + 2 * hi;
            v2f a, b;
            a.x = xk[ka];
            a.y = xk[ka + 1];
            b.x = lb[ka * D_OUT];
            b.y = lb[(ka + 1) * D_OUT];
            c = __builtin_amdgcn_wmma_f32_16x16x4_f32(
                    false, a, false, b, (short)0, c, false, false);
        }
        wait_async0();      // this wave's share of chunk kc+1 landed
        __syncthreads();    // whole chunk visible; prev buffer free to rewrite
    }

    // C/D layout: VGPR r, lanes 0-15: M=r, N=lane; lanes 16-31: M=r+8, N=lane-16.
    float* mout = m + (row0 + 8 * hi) * D_OUT + col0 + l15;
#pragma unroll
    for (int r = 0; r < 8; ++r)
        mout[r * D_OUT] = c[r];
}

// ---------------------------------------------------------------------------
// Phase 2: per-cell dot products s[i] = m[i].a_src, d[i] = m[i].a_dst
// One wave per cell; lane handles a float4 slice; wave32 shuffle reduction.
// ---------------------------------------------------------------------------
__global__ __launch_bounds__(256) void hbs_dots(const float* __restrict__ m,
                                                const float* __restrict__ avec,
                                                float* __restrict__ s,
                                                float* __restrict__ d) {
    const int lane = threadIdx.x & 31;
    const int cell = blockIdx.x * 8 + (threadIdx.x >> 5);
    if (cell >= N_CELLS) return;
    const float4 mv = ((const float4*)(m + cell * D_OUT))[lane];
    const float4 av = ((const float4*)avec)[lane];
    const float4 dv = ((const float4*)(avec + D_OUT))[lane];
    float ps = mv.x * av.x + mv.y * av.y + mv.z * av.z + mv.w * av.w;
    float pd = mv.x * dv.x + mv.y * dv.y + mv.z * dv.z + mv.w * dv.w;
#pragma unroll
    for (int off = 16; off >= 1; off >>= 1) {
        ps += __shfl_xor(ps, off, 32);
        pd += __shfl_xor(pd, off, 32);
    }
    if (lane == 0) { s[cell] = ps; d[cell] = pd; }
}

// ---------------------------------------------------------------------------
// Phase 3: per-edge leaky-relu logits + atomic row sums (L2-resident)
// ---------------------------------------------------------------------------
__global__ void hbs_edge_logits(const int* __restrict__ n_src,
                                const int* __restrict__ n_dst,
                                const float* __restrict__ s,
                                const float* __restrict__ d,
                                float* __restrict__ e,
                                float* __restrict__ row_sum) {
    int i = blockIdx.x * blockDim.x + threadIdx.x;
    if (i >= N_EDGES) return;
    const int src = n_src[i];
    float v = s[src] + d[n_dst[i]];
    v = (v > 0.0f) ? v : NEG_SLOPE * v;
    e[i] = v;
    atomicAdd(&row_sum[src], v);
}

// ---------------------------------------------------------------------------
// Phase 4: out[src] += (n_vals * e / row_sum[src]) * m[dst]
// 32 edges per wave: each lane loads one edge's scalars, then shfl-broadcast
// loop: lane handles a float4 slice -> 4x global_atomic_add_f32 (L2-resident;
// m and out both fit in the 192 MB L2). N_EDGES = 6250 * 256 exactly.
// ---------------------------------------------------------------------------
__global__ __launch_bounds__(256) void hbs_scatter(const int* __restrict__ n_src,
                                                   const int* __restrict__ n_dst,
                                                   const float* __restrict__ n_vals,
                                                   const float* __restrict__ e,
                                                   const float* __restrict__ row_sum,
                                                   const float* __restrict__ m,
                                                   float* __restrict__ out) {
    const int lane = threadIdx.x & 31;
    const int wave = threadIdx.x >> 5;
    const int eidx = (blockIdx.x * 8 + wave) * 32 + lane;  // this lane's edge
    const int   msrc = n_src[eidx];
    const int   mdst = n_dst[eidx];
    const float mw   = n_vals[eidx] * e[eidx] / row_sum[msrc];
#pragma unroll 4
    for (int i = 0; i < 32; ++i) {
        const int   src = __shfl(msrc, i, 32);
        const int   dst = __shfl(mdst, i, 32);
        const float w   = __shfl(mw,   i, 32);
        const float4 mv = ((const float4*)(m + dst * D_OUT))[lane];
        float* o = out + src * D_OUT + lane * 4;
        atomicAdd(o + 0, w * mv.x);
        atomicAdd(o + 1, w * mv.y);
        atomicAdd(o + 2, w * mv.z);
        atomicAdd(o + 3, w * mv.w);
    }
}

// ---------------------------------------------------------------------------
// Launch
// ---------------------------------------------------------------------------
extern "C" void kernel_launch(void* const* d_in, const int* in_sizes, int n_in,
                              void* d_out, int out_size, void* d_ws, size_t ws_size,
                              hipStream_t stream) {
    const float* x      = (const float*)d_in[0];
    const float* W      = (const float*)d_in[1];
    const float* avec   = (const float*)d_in[2];
    const float* n_vals = (const float*)d_in[3];
    const int*   n_src  = (const int*)d_in[4];
    const int*   n_dst  = (const int*)d_in[5];
    float* out = (float*)d_out;

    // Workspace layout (floats): m[6.4M] | e[1.6M] | row_sum[50k] | s[50k] | d[50k]
    float* ws      = (float*)d_ws;
    float* m       = ws;
    float* e       = ws + (size_t)N_CELLS * D_OUT;
    float* row_sum = e + N_EDGES;
    float* s       = row_sum + N_CELLS;
    float* d       = s + N_CELLS;

    hbs_zero<<<(N_CELLS * D_OUT + 255) / 256, 256, 0, stream>>>(out, N_CELLS * D_OUT);
    hbs_zero<<<(N_CELLS + 255) / 256, 256, 0, stream>>>(row_sum, N_CELLS);

    hbs_gemm<<<N_CELLS / 16, 256, 2 * CHUNK_BYTES, stream>>>(x, W, m);
    hbs_dots<<<(N_CELLS + 7) / 8, 256, 0, stream>>>(m, avec, s, d);
    hbs_edge_logits<<<(N_EDGES + 255) / 256, 256, 0, stream>>>(n_src, n_dst, s, d, e, row_sum);
    hbs_scatter<<<N_EDGES / (8 * 32), 256, 0, stream>>>(n_src, n_dst, n_vals, e, row_sum, m, out);
}